// SynthesisLayer_30339648979559
// MI455X (gfx1250) — compile-verified
//
#include <hip/hip_runtime.h>
#include <math.h>

// ---------------------------------------------------------------------------
// StyleGAN2 modulated conv for MI455X (gfx1250, wave32, WMMA f32 16x16x4)
//
// Decomposition (conv is linear, so modulation/demodulation can be folded):
//   s[b,i]     = w[b,:] @ style_weight[i,:] * STYLE_SCALE + style_bias[i]
//   wsq[o,i]   = sum_k conv_weight[o,i,k]^2
//   sigma[b,o] = rsqrt(CONV_SCALE^2 * sum_i s[b,i]^2 * wsq[o,i] + EPS)
//   wT[tap,i,o]= conv_weight[o,i,tap] * CONV_SCALE        (transposed once)
//   y[b,o]     = sigma[b,o] * conv2d(x[b,i]*s[b,i], wT)   (batch-SHARED weights)
//   out        = leaky_relu(y + nw[o]*noise[b] + bias[o], 0.2)
// ---------------------------------------------------------------------------

#define BATCH 16
#define CIN   512
#define COUT  512
#define HH    64
#define WW    64
#define EPSN  1e-6f

static __device__ __constant__ float STYLE_SCALE = 0.04419417382415922f;  // 1/sqrt(512)
static __device__ __constant__ float CONV_SCALE  = 0.014731391274719739f; // 1/sqrt(512*9)

typedef __attribute__((ext_vector_type(2))) float v2f;
typedef __attribute__((ext_vector_type(8))) float v8f;

// ---------------------------------------------------------------- style GEMV
__global__ void style_kernel(const float* __restrict__ w,
                             const float* __restrict__ sw,
                             const float* __restrict__ sb,
                             float* __restrict__ s) {
  int t = blockIdx.x * blockDim.x + threadIdx.x;  // 8192
  if (t >= BATCH * CIN) return;
  int b = t >> 9, i = t & 511;
  const float* wr  = w + b * 512;
  const float* swr = sw + (long)i * 512;
  float acc = 0.f;
#pragma unroll 8
  for (int j = 0; j < 512; ++j) acc += wr[j] * swr[j];
  s[t] = acc * STYLE_SCALE + sb[i];
}

// ------------------------------------------------- per-(o,i) weight sq-norm
__global__ void wsq_kernel(const float* __restrict__ cw, float* __restrict__ wsq) {
  int t = blockIdx.x * blockDim.x + threadIdx.x;  // 262144
  if (t >= COUT * CIN) return;
  const float* p = cw + (long)t * 9;
  float a = 0.f;
#pragma unroll
  for (int k = 0; k < 9; ++k) a += p[k] * p[k];
  wsq[t] = a;
}

// ------------------------------------------------------------- demodulation
__global__ void sigma_kernel(const float* __restrict__ s,
                             const float* __restrict__ wsq,
                             float* __restrict__ sigma) {
  int t = blockIdx.x * blockDim.x + threadIdx.x;  // 8192
  if (t >= BATCH * COUT) return;
  int b = t >> 9, o = t & 511;
  const float* sr = s + b * 512;
  const float* wr = wsq + (long)o * 512;
  float acc = 0.f;
#pragma unroll 4
  for (int i = 0; i < 512; ++i) { float sv = sr[i]; acc += sv * sv * wr[i]; }
  sigma[t] = rsqrtf(acc * CONV_SCALE * CONV_SCALE + EPSN);
}

// ----------------------------------------- transpose weights to [tap][ci][o]
__global__ void wprep_kernel(const float* __restrict__ cw, float* __restrict__ wT) {
  long t = (long)blockIdx.x * blockDim.x + threadIdx.x;  // 9*512*512
  if (t >= (long)9 * CIN * COUT) return;
  int o    = (int)(t & 511);
  long r   = t >> 9;
  int ci   = (int)(r & 511);
  int tap  = (int)(r >> 9);
  wT[t] = cw[((long)o * CIN + ci) * 9 + tap] * CONV_SCALE;
}

// --------------------------------------------------------- main WMMA conv
// Block tile: 128 out-ch x 128 spatial (2 image rows). 8 waves; each wave
// computes a 32x64 sub-tile = 2x4 WMMA 16x16 accumulators (v8f each).
#define OC_TILE 128
#define CI_TILE 32
#define ROWS    2

__launch_bounds__(256)
__global__ void modconv_kernel(const float* __restrict__ x,
                               const float* __restrict__ wT,
                               const float* __restrict__ s,
                               const float* __restrict__ sigma,
                               const float* __restrict__ nw,
                               const float* __restrict__ bias,
                               const float* __restrict__ noise,
                               float* __restrict__ out) {
  __shared__ float lds_in[CI_TILE][ROWS + 2][WW + 2];      // haloed input, pre-modulated
  __shared__ __align__(16) float lds_w[CI_TILE][OC_TILE];  // weight tap tile [k][o]
  __shared__ float lds_s[CI_TILE];

  const int b    = blockIdx.z;
  const int oc0  = blockIdx.y * OC_TILE;
  const int h0   = blockIdx.x * ROWS;
  const int tid  = threadIdx.x;
  const int lane = tid & 31;
  const int wid  = tid >> 5;            // 0..7
  const int m_base = (wid & 3) * 32;    // out-ch offset inside tile
  const int n_base = (wid >> 2) * 64;   // spatial offset inside tile
  const int lanelo = lane & 15;
  const int khalf  = (lane >> 4) * 2;   // ISA 16x4 f32 A/B split: lanes 16-31 hold K+2
  const int mhalf  = (lane >> 4) * 8;   // C/D layout: lanes 16-31 hold M+8

  v8f acc[2][4];
#pragma unroll
  for (int tm = 0; tm < 2; ++tm)
#pragma unroll
    for (int tn = 0; tn < 4; ++tn) { v8f z = {}; acc[tm][tn] = z; }

  for (int cc = 0; cc < CIN; cc += CI_TILE) {
    __syncthreads();  // protect lds_in / lds_s from previous iteration's readers
    if (tid < CI_TILE) lds_s[tid] = s[b * CIN + cc + tid];
    __syncthreads();

    // Stage haloed, style-modulated input tile: 32ch x 4rows x 66cols = 8448 elems
    for (int e = tid; e < CI_TILE * (ROWS + 2) * (WW + 2); e += 256) {
      int ch  = e / ((ROWS + 2) * (WW + 2));
      int rem = e - ch * ((ROWS + 2) * (WW + 2));
      int r   = rem / (WW + 2);
      int col = rem - r * (WW + 2);
      int h   = h0 - 1 + r;
      int wc  = col - 1;
      float v = 0.f;
      if (h >= 0 && h < HH && wc >= 0 && wc < WW)
        v = x[(((long)b * CIN + cc + ch) * HH + h) * WW + wc] * lds_s[ch];
      lds_in[ch][r][col] = v;
    }

    for (int tap = 0; tap < 9; ++tap) {
      __syncthreads();  // previous lds_w readers done; lds_in writes visible (tap 0)
      // Stage weight tap tile (coalesced float4 from the transposed wT buffer)
      for (int e = tid; e < CI_TILE * OC_TILE / 4; e += 256) {
        int k = e >> 5;           // (e*4)/128
        int o = (e << 2) & 127;
        float4 v = *(const float4*)&wT[((long)tap * CIN + cc + k) * COUT + oc0 + o];
        *(float4*)&lds_w[k][o] = v;
      }
      __syncthreads();

      const int kh = tap / 3, kw = tap - kh * 3;
#pragma unroll
      for (int k0 = 0; k0 < CI_TILE; k0 += 4) {
        v2f afrag[2], bfrag[4];
#pragma unroll
        for (int tm = 0; tm < 2; ++tm) {
          int m = m_base + tm * 16 + lanelo;
          afrag[tm].x = lds_w[k0 + khalf    ][m];
          afrag[tm].y = lds_w[k0 + khalf + 1][m];
        }
#pragma unroll
        for (int tn = 0; tn < 4; ++tn) {
          int n  = n_base + tn * 16 + lanelo;
          int r  = (n >> 6) + kh;          // image row within haloed tile
          int wc = (n & 63) + kw;          // 0..65 (halo handles kw shift)
          bfrag[tn].x = lds_in[k0 + khalf    ][r][wc];
          bfrag[tn].y = lds_in[k0 + khalf + 1][r][wc];
        }
#pragma unroll
        for (int tm = 0; tm < 2; ++tm)
#pragma unroll
          for (int tn = 0; tn < 4; ++tn)
            acc[tm][tn] = __builtin_amdgcn_wmma_f32_16x16x4_f32(
                false, afrag[tm], false, bfrag[tn],
                (short)0, acc[tm][tn], false, false);
      }
    }
  }

  // Epilogue: demodulate, noise, bias, leaky ReLU (0.2)
#pragma unroll
  for (int tm = 0; tm < 2; ++tm) {
#pragma unroll
    for (int tn = 0; tn < 4; ++tn) {
      int n  = n_base + tn * 16 + lanelo;
      int r  = n >> 6;
      int wc = n & 63;
      int h  = h0 + r;
      float nz = noise[((long)b * HH + h) * WW + wc];
#pragma unroll
      for (int v = 0; v < 8; ++v) {
        int o = oc0 + m_base + tm * 16 + mhalf + v;
        float val = acc[tm][tn][v] * sigma[b * COUT + o] + nw[o] * nz + bias[o];
        val = (val > 0.f) ? val : 0.2f * val;
        out[(((long)b * COUT + o) * HH + h) * WW + wc] = val;
      }
    }
  }
}

// ---------------------------------------------------------------------------
extern "C" void kernel_launch(void* const* d_in, const int* in_sizes, int n_in,
                              void* d_out, int out_size, void* d_ws, size_t ws_size,
                              hipStream_t stream) {
  const float* x     = (const float*)d_in[0];
  const float* w     = (const float*)d_in[1];
  const float* sw    = (const float*)d_in[2];
  const float* sb    = (const float*)d_in[3];
  const float* cw    = (const float*)d_in[4];
  const float* nw    = (const float*)d_in[5];
  const float* bias  = (const float*)d_in[6];
  const float* noise = (const float*)d_in[7];
  float* out = (float*)d_out;

  float* ws    = (float*)d_ws;
  float* s_buf = ws;                            //  16*512
  float* wsq   = s_buf + BATCH * CIN;           // 512*512
  float* sigma = wsq + COUT * CIN;              //  16*512
  float* wT    = sigma + BATCH * COUT;          // 9*512*512  (~10.6 MB total)

  style_kernel<<<(BATCH * CIN + 255) / 256, 256, 0, stream>>>(w, sw, sb, s_buf);
  wsq_kernel  <<<(COUT * CIN + 255) / 256, 256, 0, stream>>>(cw, wsq);
  sigma_kernel<<<(BATCH * COUT + 255) / 256, 256, 0, stream>>>(s_buf, wsq, sigma);
  wprep_kernel<<<(9 * CIN * COUT + 255) / 256, 256, 0, stream>>>(cw, wT);

  dim3 grid(HH / ROWS, COUT / OC_TILE, BATCH);  // 32 x 4 x 16
  modconv_kernel<<<grid, 256, 0, stream>>>(x, wT, s_buf, sigma, nw, bias, noise, out);
}